// DynamicConv1dTBC_86449101734001
// MI455X (gfx1250) — compile-verified
//
#include <hip/hip_runtime.h>
#include <math.h>

// Problem constants (match reference)
#define TT 2048
#define BB 4
#define CC 512
#define HH 16
#define KK 31
#define PP 15
#define MM (TT * BB)   // 8192 rows

typedef __attribute__((ext_vector_type(2))) float v2f;
typedef __attribute__((ext_vector_type(8))) float v8f;

// ---------------- fp32 WMMA GEMM: C[m,n] = sum_k A[m,k] * W[n,k] (+ bias[n]) ----
// A: row-major M x K (lda = K). W: row-major N x K (we multiply by W^T).
// Block tile 64(M) x 128(N), 128 threads = 4 waves; each wave computes a 16x128
// strip as 8 16x16 fragments. K staged through double-buffered LDS, 32 per panel.
#define TM 64
#define TN 128
#define KC 32
#define ASTRIDE 36          // floats per A row in LDS (16B-aligned, conflict-free)
#define BSTRIDE (TN + 16)   // float2 per B k-pair row (row stride == 32 floats mod 64)

__global__ __launch_bounds__(128) void gemm_f32_wmma(
    const float* __restrict__ A, const float* __restrict__ W,
    const float* __restrict__ bias, float* __restrict__ Co,
    int M, int N_valid, int K, int ldc)
{
    __shared__ float As[2][TM][ASTRIDE];       // A panel: 64 x 32 (padded)
    __shared__ v2f   Bs[2][KC / 2][BSTRIDE];   // B panel as K-pairs: Bs[g][n] = {Wk(2g), Wk(2g+1)}

    const int tid  = threadIdx.x;      // 0..127
    const int wave = tid >> 5;         // 0..3
    const int lane = tid & 31;
    const int hl   = lane >> 4;        // half-lane group: K offset 0 or 2
    const int l15  = lane & 15;

    const int m0   = blockIdx.x * TM;
    const int n0   = blockIdx.y * TN;
    const int mrow = wave * 16;

    // fill mappings (lane-minor in the LDS-written dimension -> conflict-free stores)
    const int rowA = tid & 15;         // + r*16, r=0..3
    const int c4A  = tid >> 4;         // 0..7 -> k chunk of 4
    const int nB   = tid & 15;         // + r*16, r=0..7
    const int c4B  = tid >> 4;         // 0..7 -> k chunk of 4

    v8f acc[8] = {};

    // ---- preload panel 0 ----
    {
        #pragma unroll
        for (int r = 0; r < TM / 16; r++) {
            const float4 v = *(const float4*)(A + (size_t)(m0 + rowA + r * 16) * K + c4A * 4);
            *((float4*)&As[0][rowA + r * 16][c4A * 4]) = v;
        }
        #pragma unroll
        for (int r = 0; r < TN / 16; r++) {
            const int gn = n0 + nB + r * 16;
            float4 v = make_float4(0.f, 0.f, 0.f, 0.f);
            if (gn < N_valid) v = *(const float4*)(W + (size_t)gn * K + c4B * 4);
            Bs[0][c4B * 2 + 0][nB + r * 16] = v2f{v.x, v.y};
            Bs[0][c4B * 2 + 1][nB + r * 16] = v2f{v.z, v.w};
        }
    }
    __syncthreads();

    for (int kc = 0; kc < K; kc += KC) {
        const int cur = (kc / KC) & 1;
        const int nxt = cur ^ 1;
        const bool has_next = (kc + KC) < K;

        // ---- issue next panel's global loads into registers ----
        float4 ra[TM / 16];
        float4 rb[TN / 16];
        if (has_next) {
            #pragma unroll
            for (int r = 0; r < TM / 16; r++)
                ra[r] = *(const float4*)(A + (size_t)(m0 + rowA + r * 16) * K + kc + KC + c4A * 4);
            #pragma unroll
            for (int r = 0; r < TN / 16; r++) {
                const int gn = n0 + nB + r * 16;
                rb[r] = make_float4(0.f, 0.f, 0.f, 0.f);
                if (gn < N_valid)
                    rb[r] = *(const float4*)(W + (size_t)gn * K + kc + KC + c4B * 4);
            }
        }

        // ---- compute current panel: 8 k-steps x 8 fragments = 64 WMMA ----
        #pragma unroll
        for (int kk = 0; kk < KC; kk += 4) {
            const int ka = kk + hl * 2;        // lanes 0-15: kk,kk+1 ; lanes 16-31: kk+2,kk+3
            const int g  = (kk >> 1) + hl;     // B float2 k-pair row
            v2f a;
            a.x = As[cur][mrow + l15][ka];
            a.y = As[cur][mrow + l15][ka + 1];
            #pragma unroll
            for (int j = 0; j < 8; j++) {
                const v2f b = Bs[cur][g][j * 16 + l15];   // single ds_load_b64
                acc[j] = __builtin_amdgcn_wmma_f32_16x16x4_f32(
                    false, a, false, b, (short)0, acc[j], false, false);
            }
        }

        // ---- commit next panel to LDS, then one barrier per iteration ----
        if (has_next) {
            #pragma unroll
            for (int r = 0; r < TM / 16; r++)
                *((float4*)&As[nxt][rowA + r * 16][c4A * 4]) = ra[r];
            #pragma unroll
            for (int r = 0; r < TN / 16; r++) {
                Bs[nxt][c4B * 2 + 0][nB + r * 16] = v2f{rb[r].x, rb[r].y};
                Bs[nxt][c4B * 2 + 1][nB + r * 16] = v2f{rb[r].z, rb[r].w};
            }
        }
        __syncthreads();
    }

    // ---- store: C/D layout -> lanes 0-15: M = i (VGPR i), lanes 16-31: M = 8+i ----
    const int rbase = m0 + mrow + hl * 8;
    #pragma unroll
    for (int j = 0; j < 8; j++) {
        const int col = n0 + j * 16 + l15;
        if (col < N_valid) {
            const float bv = bias ? bias[col] : 0.0f;
            #pragma unroll
            for (int i = 0; i < 8; i++)
                Co[(size_t)(rbase + i) * ldc + col] = acc[j][i] + bv;
        }
    }
}

// ---------------- softmax over K=31 per (row, head), in place -------------------
__global__ __launch_bounds__(256) void softmax_k31(float* __restrict__ w, int M)
{
    const int idx = blockIdx.x * blockDim.x + threadIdx.x;   // m*HH + head
    if (idx >= M * HH) return;
    const int m = idx >> 4;
    const int head = idx & (HH - 1);
    float* p = w + (size_t)m * CC + head * KK;   // row stride padded to CC
    float mx = -INFINITY;
    #pragma unroll
    for (int k = 0; k < KK; k++) mx = fmaxf(mx, p[k]);
    float s = 0.0f;
    #pragma unroll
    for (int k = 0; k < KK; k++) { const float e = __expf(p[k] - mx); p[k] = e; s += e; }
    const float inv = 1.0f / s;
    #pragma unroll
    for (int k = 0; k < KK; k++) p[k] *= inv;
}

// ---------------- dynamic banded convolution ------------------------------------
// out[t,b,c] = sum_k h[t-P+k, b, c] * wsm[t,b,head(c),k]
__global__ __launch_bounds__(256) void dynconv(
    const float* __restrict__ h, const float* __restrict__ w, float* __restrict__ out)
{
    __shared__ float ws[HH][KK];
    const int tb = blockIdx.x;           // t*BB + b
    const int t = tb / BB;
    const int b = tb % BB;
    const int tid = threadIdx.x;

    for (int i = tid; i < HH * KK; i += 256)
        ws[i / KK][i % KK] = w[(size_t)tb * CC + i];
    __syncthreads();

    #pragma unroll
    for (int c = tid; c < CC; c += 256) {
        const int head = c >> 5;         // R = 32 channels per head
        float acc = 0.0f;
        #pragma unroll
        for (int k = 0; k < KK; k++) {
            const int ts = t - PP + k;
            if (ts >= 0 && ts < TT)
                acc = fmaf(h[((size_t)ts * BB + b) * CC + c], ws[head][k], acc);
        }
        out[(size_t)tb * CC + c] = acc;
    }
}

extern "C" void kernel_launch(void* const* d_in, const int* in_sizes, int n_in,
                              void* d_out, int out_size, void* d_ws, size_t ws_size,
                              hipStream_t stream) {
    const float* x  = (const float*)d_in[0];   // (T,B,C)
    const float* W1 = (const float*)d_in[1];   // (C,C)
    const float* b1 = (const float*)d_in[2];   // (C,)
    const float* Wl = (const float*)d_in[3];   // (H*K, C)
    const float* W2 = (const float*)d_in[4];   // (C,C)
    const float* b2 = (const float*)d_in[5];   // (C,)
    float* y = (float*)d_out;

    float* h  = (float*)d_ws;                  // 8192 x 512
    float* w  = h  + (size_t)MM * CC;          // 8192 x 512 (496 valid cols)
    float* cv = w  + (size_t)MM * CC;          // 8192 x 512

    const dim3 block(128);
    const dim3 grid(MM / TM, CC / TN);         // 128 x 4

    // 1) h = x @ W1^T + b1
    gemm_f32_wmma<<<grid, block, 0, stream>>>(x, W1, b1, h, MM, CC, CC, CC);
    // 2) w = h @ Wl^T   (N_valid = H*K = 496, buffer padded to 512)
    gemm_f32_wmma<<<grid, block, 0, stream>>>(h, Wl, nullptr, w, MM, HH * KK, CC, CC);
    // 3) softmax over K per (m, head)
    softmax_k31<<<(MM * HH + 255) / 256, 256, 0, stream>>>(w, MM);
    // 4) dynamic conv
    dynconv<<<MM, 256, 0, stream>>>(h, w, cv);
    // 5) y = conv @ W2^T + b2
    gemm_f32_wmma<<<grid, block, 0, stream>>>(cv, W2, b2, y, MM, CC, CC, CC);
}